// ScatterImpl_2954937499912
// MI455X (gfx1250) — compile-verified
//
#include <hip/hip_runtime.h>
#include <stdint.h>

// segment_sum (scatter-add, reduce='sum') for src[E,64] f32, index[E] i64 -> out[100k,64] f32.
// Memory-bound: ~450MB total traffic, roofline ~20us @ 23.3 TB/s.
// CDNA5 paths used: global_load_async_to_lds_b128 (double-buffered, NT hint),
// global_store_async_from_lds_b128 (zero-fill), s_wait_asynccnt, L2-resident f32 atomics.

#define TPB 256         // 8 wave32 per block
#define QPE 16          // float4 quarters per 64-float edge row
#define EPT (TPB / QPE) // 16 edges per tile; tile = 256 float4 = 4KB

__device__ __forceinline__ void issue_async_tile(const float4* g, uint32_t lds_off) {
  // per-lane: copies 16 bytes from global address g to the LDS address in lds_off, tracked on ASYNCcnt
  asm volatile("global_load_async_to_lds_b128 %0, %1, off th:TH_LOAD_NT"
               :: "v"(lds_off), "v"((uint64_t)(uintptr_t)g)
               : "memory");
}

__device__ __forceinline__ void scatter_quarters(float4 v, long long seg, int sub,
                                                 float* __restrict__ out) {
  float* dst = out + seg * 64 + (sub << 2);
  // no-return f32 atomic adds -> RMW resolved in (192MB) L2 where out is resident
  __hip_atomic_fetch_add(dst + 0, v.x, __ATOMIC_RELAXED, __HIP_MEMORY_SCOPE_AGENT);
  __hip_atomic_fetch_add(dst + 1, v.y, __ATOMIC_RELAXED, __HIP_MEMORY_SCOPE_AGENT);
  __hip_atomic_fetch_add(dst + 2, v.z, __ATOMIC_RELAXED, __HIP_MEMORY_SCOPE_AGENT);
  __hip_atomic_fetch_add(dst + 3, v.w, __ATOMIC_RELAXED, __HIP_MEMORY_SCOPE_AGENT);
}

// Zero d_out by streaming zeros from a single LDS slot with async stores.
__global__ void zero_out_async_kernel(float* __restrict__ out, int n4) {
  __shared__ float4 zslot;
  if (threadIdx.x == 0) zslot = make_float4(0.f, 0.f, 0.f, 0.f);
  __syncthreads();  // LDS write committed before the DMA engine reads it
  uint32_t lds_off = (uint32_t)(uintptr_t)&zslot;
  int stride = (int)(gridDim.x * blockDim.x);
  for (int i = (int)(blockIdx.x * blockDim.x + threadIdx.x); i < n4; i += stride) {
    const float4* dst = (const float4*)out + i;
    asm volatile("global_store_async_from_lds_b128 %0, %1, off"
                 :: "v"((uint64_t)(uintptr_t)dst), "v"(lds_off)
                 : "memory");
  }
  asm volatile("s_wait_asynccnt 0x0" ::: "memory");
}

// Main scatter: grid-stride over 16-edge tiles, double-buffered async staging.
// Each lane owns its 16B slot (async write + ds read by same lane -> no barriers).
__global__ void scatter_add_async_kernel(const float4* __restrict__ src4,
                                         const long long* __restrict__ index,
                                         float* __restrict__ out, int tiles) {
  __shared__ float4 stage[2][TPB];  // 8KB
  const int t   = (int)threadIdx.x;
  const int sub = t & (QPE - 1);  // which float4 of the row
  const int er  = t >> 4;         // which edge in the tile
  const uint32_t lds0 = (uint32_t)(uintptr_t)&stage[0][t];
  const uint32_t lds1 = (uint32_t)(uintptr_t)&stage[1][t];
  const int stride = (int)gridDim.x;

  int tile = (int)blockIdx.x;
  if (tile >= tiles) return;
  issue_async_tile(src4 + (size_t)tile * TPB + t, lds0);

  for (;;) {
    // ---- buffer 0 holds `tile` ----
    int next = tile + stride;
    if (next < tiles) {
      issue_async_tile(src4 + (size_t)next * TPB + t, lds1);
      asm volatile("s_wait_asynccnt 0x1" ::: "memory");  // oldest (buf0) done
    } else {
      asm volatile("s_wait_asynccnt 0x0" ::: "memory");
    }
    {
      float4 v = stage[0][t];
      long long seg =
          __builtin_nontemporal_load(index + (long long)tile * EPT + er);
      scatter_quarters(v, seg, sub, out);
    }
    if (next >= tiles) break;
    tile = next;

    // ---- buffer 1 holds `tile` ----
    next = tile + stride;
    if (next < tiles) {
      issue_async_tile(src4 + (size_t)next * TPB + t, lds0);
      asm volatile("s_wait_asynccnt 0x1" ::: "memory");
    } else {
      asm volatile("s_wait_asynccnt 0x0" ::: "memory");
    }
    {
      float4 v = stage[1][t];
      long long seg =
          __builtin_nontemporal_load(index + (long long)tile * EPT + er);
      scatter_quarters(v, seg, sub, out);
    }
    if (next >= tiles) break;
    tile = next;
  }
}

// Generic tail (E not divisible by 16) — plain path, one thread per float4.
__global__ void scatter_tail_kernel(const float4* __restrict__ src4,
                                    const long long* __restrict__ index,
                                    float* __restrict__ out,
                                    int startEdge, int E) {
  int i = (int)(blockIdx.x * blockDim.x + threadIdx.x);
  int e = startEdge + (i >> 4);
  if (e >= E) return;
  int sub = i & (QPE - 1);
  float4 v = src4[(size_t)e * QPE + sub];
  long long seg = index[e];
  scatter_quarters(v, seg, sub, out);
}

extern "C" void kernel_launch(void* const* d_in, const int* in_sizes, int n_in,
                              void* d_out, int out_size, void* d_ws, size_t ws_size,
                              hipStream_t stream) {
  (void)n_in; (void)d_ws; (void)ws_size;
  const int F = 64;
  const float4*    src4  = (const float4*)d_in[0];
  const long long* index = (const long long*)d_in[1];  // jax int64
  float*           out   = (float*)d_out;
  const int E  = in_sizes[0] / F;
  const int n4 = out_size / 4;

  // 1) zero the output (required every call: accumulation + poisoned buffer)
  int zblocks = (n4 + TPB - 1) / TPB;
  if (zblocks > 4096) zblocks = 4096;
  zero_out_async_kernel<<<zblocks, TPB, 0, stream>>>(out, n4);

  // 2) scatter-add the bulk in 16-edge tiles
  const int tiles = E / EPT;
  if (tiles > 0) {
    int blocks = tiles < 8192 ? tiles : 8192;
    scatter_add_async_kernel<<<blocks, TPB, 0, stream>>>(src4, index, out, tiles);
  }

  // 3) tail edges, if any
  const int tail = E - tiles * EPT;
  if (tail > 0) {
    int n = tail * QPE;
    scatter_tail_kernel<<<(n + TPB - 1) / TPB, TPB, 0, stream>>>(
        src4, index, out, tiles * EPT, E);
  }
}